// ChamferDistanceL2_32839319945864
// MI455X (gfx1250) — compile-verified
//
#include <hip/hip_runtime.h>
#include <math.h>

typedef __attribute__((ext_vector_type(2))) float v2f;
typedef __attribute__((ext_vector_type(8))) float v8f;

#define BATCH   16
#define NPTS    4096            // N == M
#define STRIPS_PER_BATCH (NPTS / 16)   // 256
#define TOTAL_WAVES (BATCH * STRIPS_PER_BATCH) // 4096
#define COL_BLOCKS (NPTS / 16)  // 256

// ---------------------------------------------------------------------------
// Pack kernel: build WMMA-ready operand arrays for both directions.
//   "A-pack":  (-2x, -2y, -2z, |p|^2)   -> used as the COLUMN (B) operand
//   "B-pack":  ( x,   y,   z,  1    )   -> used as the ROW (A) operand
//   norm arrays feed the loop-invariant C-matrix (row norms).
// D[m,n] = (p_m . 1-pack) x (-2g_n, |g_n|^2) + |p_m|^2
//        = |p_m|^2 - 2 p_m.g_n + |g_n|^2     (exact squared distance)
// ---------------------------------------------------------------------------
__global__ __launch_bounds__(256)
void chamfer_pack_kernel(const float* __restrict__ pred,
                         const float* __restrict__ gt,
                         float4* __restrict__ predA, float4* __restrict__ predB,
                         float4* __restrict__ gtA,   float4* __restrict__ gtB,
                         float*  __restrict__ normP, float* __restrict__ normG,
                         int total)
{
    int i = blockIdx.x * blockDim.x + threadIdx.x;
    if (i >= total) return;

    float px = pred[3 * i + 0];
    float py = pred[3 * i + 1];
    float pz = pred[3 * i + 2];
    float pn = px * px + py * py + pz * pz;
    predA[i] = make_float4(-2.0f * px, -2.0f * py, -2.0f * pz, pn);
    predB[i] = make_float4(px, py, pz, 1.0f);
    normP[i] = pn;

    float gx = gt[3 * i + 0];
    float gy = gt[3 * i + 1];
    float gz = gt[3 * i + 2];
    float gn = gx * gx + gy * gy + gz * gz;
    gtA[i] = make_float4(-2.0f * gx, -2.0f * gy, -2.0f * gz, gn);
    gtB[i] = make_float4(gx, gy, gz, 1.0f);
    normG[i] = gn;
}

// ---------------------------------------------------------------------------
// Directional min kernel: each wave owns a 16-row strip of the distance
// matrix for one batch and sweeps all 16-column blocks with
// V_WMMA_F32_16X16X4_F32.
//   A (rows)   : (x, y, z, 1)          -- loop invariant, 2 VGPRs
//   B (cols)   : (-2g, |g|^2)          -- one b64 load per tile
//   C (rows)   : |p_m|^2 broadcast     -- loop invariant, 8 VGPRs
// Inner loop = 1 load + 1 WMMA + 8 mins. Row-wise running min lives in 8
// accumulator VGPRs per lane (C/D layout: VGPR r = row r for lanes 0-15,
// row r+8 for lanes 16-31).
// ---------------------------------------------------------------------------
__global__ __launch_bounds__(256)
void chamfer_dir_kernel(const float4* __restrict__ rowPts,  // [B*NPTS] (x,y,z,1) rows
                        const float4* __restrict__ colPack, // [B*NPTS] (-2g,|g|^2) cols
                        const float*  __restrict__ rowNorm, // [B*NPTS] |row|^2
                        float* __restrict__ outMin)         // [B*NPTS] per-row min
{
    int gwave = (blockIdx.x * 256 + (int)threadIdx.x) >> 5;   // 0..4095
    int lane  = threadIdx.x & 31;
    int rlane = lane & 15;
    int half  = lane >> 4;

    int batch = gwave >> 8;            // 256 strips per batch
    int row0  = (gwave & 255) << 4;    // strip base row
    int base  = batch << 12;           // batch * 4096

    // A operand: lane<16 holds (K=0,K=1)=(x,y) of row rlane; lane>=16 holds
    // (K=2,K=3)=(z,1). Pack is (x,y,z,1), so take the half-th float2.
    const float2* ap = (const float2*)(rowPts + base + row0 + rlane);
    float2 av = ap[half];
    v2f a; a[0] = av.x; a[1] = av.y;

    // Loop-invariant C operand: VGPR r holds |p|^2 of row (r + 8*half).
    const float* nrm = rowNorm + base + row0 + half * 8;
    v8f c;
#pragma unroll
    for (int i = 0; i < 8; ++i) c[i] = nrm[i];

    float acc[8];
#pragma unroll
    for (int i = 0; i < 8; ++i) acc[i] = INFINITY;

    for (int cb = 0; cb < COL_BLOCKS; ++cb) {
        int cidx = base + (cb << 4) + rlane;

        // B operand: lane<16 holds (-2gx,-2gy) of column rlane;
        // lane>=16 holds (-2gz, |g|^2).
        const float2* bp = (const float2*)(colPack + cidx);
        float2 bv = bp[half];
        v2f b; b[0] = bv.x; b[1] = bv.y;

        // D[m,n] = |p_m|^2 - 2 p_m.g_n + |g_n|^2
        v8f d = __builtin_amdgcn_wmma_f32_16x16x4_f32(
            /*neg_a=*/false, a, /*neg_b=*/false, b,
            /*c_mod=*/(short)0, c, /*reuse_a=*/false, /*reuse_b=*/false);

#pragma unroll
        for (int i = 0; i < 8; ++i) acc[i] = fminf(acc[i], d[i]);
    }

    // Row-min: reduce across the 16 lanes of each half independently
    // (lanes 0-15 cover cols n=0..15 of rows r, lanes 16-31 of rows r+8).
#pragma unroll
    for (int i = 0; i < 8; ++i) {
        float v = acc[i];
        v = fminf(v, __shfl_xor(v, 1, 16));
        v = fminf(v, __shfl_xor(v, 2, 16));
        v = fminf(v, __shfl_xor(v, 4, 16));
        v = fminf(v, __shfl_xor(v, 8, 16));
        acc[i] = v;
    }

    int rowOff = half * 8;
#pragma unroll
    for (int r = 0; r < 8; ++r) {
        if (rlane == r) outMin[base + row0 + r + rowOff] = acc[r];
    }
}

// ---------------------------------------------------------------------------
// Final reduction: mean(dist1) + mean(dist2) = (sum over both buffers)/65536.
// Single workgroup, LDS tree reduce.
// ---------------------------------------------------------------------------
__global__ __launch_bounds__(256)
void chamfer_reduce_kernel(const float* __restrict__ dists, float* __restrict__ out,
                           int n, float invMeanN)
{
    __shared__ float sm[256];
    float s = 0.0f;
    for (int i = threadIdx.x; i < n; i += 256) s += dists[i];
    sm[threadIdx.x] = s;
    __syncthreads();
    for (int w = 128; w > 0; w >>= 1) {
        if ((int)threadIdx.x < w) sm[threadIdx.x] += sm[threadIdx.x + w];
        __syncthreads();
    }
    if (threadIdx.x == 0) out[0] = sm[0] * invMeanN;
}

extern "C" void kernel_launch(void* const* d_in, const int* in_sizes, int n_in,
                              void* d_out, int out_size, void* d_ws, size_t ws_size,
                              hipStream_t stream) {
    const float* pred = (const float*)d_in[0];   // [16,4096,3]
    const float* gt   = (const float*)d_in[1];   // [16,4096,3]
    float* out = (float*)d_out;

    const int total = BATCH * NPTS;              // 65536 points per cloud set

    // Workspace carve-up (floats). ~5.2 MB total.
    float* ws = (float*)d_ws;
    float4* predA = (float4*)(ws + 0);           // 262144 floats
    float4* predB = (float4*)(ws + 262144);
    float4* gtA   = (float4*)(ws + 524288);
    float4* gtB   = (float4*)(ws + 786432);
    float*  normP = ws + 1048576;                // 65536
    float*  normG = ws + 1114112;                // 65536
    float*  dist1 = ws + 1179648;                // 65536
    float*  dist2 = ws + 1245184;                // 65536 (contiguous after dist1)

    chamfer_pack_kernel<<<total / 256, 256, 0, stream>>>(
        pred, gt, predA, predB, gtA, gtB, normP, normG, total);

    // dist1: rows = pred (x,y,z,1 / |p|^2), cols = gt (-2g,|g|^2)
    chamfer_dir_kernel<<<(TOTAL_WAVES * 32) / 256, 256, 0, stream>>>(
        predB, gtA, normP, dist1);

    // dist2: rows = gt, cols = pred
    chamfer_dir_kernel<<<(TOTAL_WAVES * 32) / 256, 256, 0, stream>>>(
        gtB, predA, normG, dist2);

    chamfer_reduce_kernel<<<1, 256, 0, stream>>>(
        dist1, out, 2 * total, 1.0f / (float)total);
}